// GaussianRenderer_56049323213574
// MI455X (gfx1250) — compile-verified
//
#include <hip/hip_runtime.h>
#include <hip/hip_bf16.h>
#include <stdint.h>

typedef __attribute__((ext_vector_type(2))) float v2f;
typedef __attribute__((ext_vector_type(8))) float v8f;

#define NBATCH 2
#define NGAUSS 768
#define IMG    160
#define STRIDEF 12                         /* floats per gaussian record (48 B) */
#define LDSBYTES (NGAUSS * STRIDEF * 4)    /* 36864 B */
#define STRIPS_PER_IMG ((IMG * IMG) / 16)  /* 1600 16-pixel strips */
#define WAVES_PER_BLOCK 8
#define BLOCKS_PER_IMG (STRIPS_PER_IMG / WAVES_PER_BLOCK) /* 200 */

/* Raw v_exp_f32: no denorm slow-path. Exponents < -126 mean alpha ~ 1e-38,
   i.e. zero contribution, so flush-to-zero is exactly what we want. */
#if __has_builtin(__builtin_amdgcn_exp2f)
#define EXP2F(x) __builtin_amdgcn_exp2f(x)
#else
#define EXP2F(x) exp2f(x)
#endif

/* Single v_med3_f32 clamp (no canonicalize pair from fminf/fmaxf). */
#if __has_builtin(__builtin_amdgcn_fmed3f)
#define CLAMPF(x, lo, hi) __builtin_amdgcn_fmed3f((x), (lo), (hi))
#else
#define CLAMPF(x, lo, hi) fminf(fmaxf((x), (lo)), (hi))
#endif

// ---------------------------------------------------------------------------
// Kernel 1: per-gaussian preprocessing.
// Record layout (floats):
//   [0..3]  = K0,K1,K4,K5   (A operand pair for WMMA#1/#2, low lanes)
//   [4..7]  = K2,K3,0,0     (A operand pair for WMMA#1/#2, high lanes)
//   [8..11] = r,g,b,0
// s2(p,g) = log2e*(-q/2) + log2(op) = f(p) . K(g) with features
//   f = [gx^2, gy^2, gx*gy, gx, gy, 1]  (coords centered at image center)
// ---------------------------------------------------------------------------
__global__ __launch_bounds__(256) void gs_prep(
    const float* __restrict__ means, const float* __restrict__ scales,
    const float* __restrict__ rots, const float* __restrict__ colors,
    const float* __restrict__ opac, float* __restrict__ ws) {
  int i = blockIdx.x * 256 + threadIdx.x;
  if (i >= NBATCH * NGAUSS) return;
  float mx = means[i * 2 + 0], my = means[i * 2 + 1];
  float sx = scales[i * 2 + 0], sy = scales[i * 2 + 1];
  float th = rots[i];
  float c = cosf(th), s = sinf(th);
  float sx2 = sx * sx, sy2 = sy * sy;
  float a = c * c * sx2 + s * s * sy2;
  float b = c * s * (sx2 - sy2);
  float d = s * s * sx2 + c * c * sy2;
  const float SZ2 = 1.0e-6f;   /* 0.001^2 */
  const float EPS2D = 0.3f;
  float cxx = a + SZ2 * mx * mx + EPS2D;
  float cxy = b + SZ2 * mx * my;
  float cyy = d + SZ2 * my * my + EPS2D;
  float det = cxx * cyy - cxy * cxy;
  float ca = cyy / det, cb = -cxy / det, cc = cxx / det;
  float cx = mx - 0.5f * (float)IMG;       /* center coords for conditioning */
  float cy = my - 0.5f * (float)IMG;
  const float L = 1.4426950408889634f;     /* log2(e) */
  float K0 = -0.5f * L * ca;
  float K1 = -0.5f * L * cc;
  float K2 = -L * cb;
  float K3 = L * (ca * cx + cb * cy);
  float K4 = L * (cc * cy + cb * cx);
  float K5 = -0.5f * L * (ca * cx * cx + 2.0f * cb * cx * cy + cc * cy * cy)
             + log2f(opac[i]);             /* fold opacity into exponent */
  float* o = ws + (size_t)i * STRIDEF;
  o[0] = K0; o[1] = K1; o[2]  = K4;               o[3]  = K5;
  o[4] = K2; o[5] = K3; o[6]  = 0.0f;             o[7]  = 0.0f;
  o[8] = colors[i * 3 + 0]; o[9] = colors[i * 3 + 1];
  o[10] = colors[i * 3 + 2]; o[11] = 0.0f;
}

// ---------------------------------------------------------------------------
// Kernel 2: render. One wave = one 16x1 pixel strip (pixel state duplicated in
// lane pairs l / l+16). Per 16-gaussian tile: 2x v_wmma_f32_16x16x4_f32 give
// the 16x16 exponent tile; each lane composites its 8 in-order gaussians, then
// a half-wave shuffle merges the two ordered segments.
// ---------------------------------------------------------------------------
__global__ __launch_bounds__(256) void gs_render(const float* __restrict__ ws,
                                                 float* __restrict__ out) {
  __shared__ float sG[NGAUSS * STRIDEF];

  const int tid = threadIdx.x;
  const int batch = blockIdx.x / BLOCKS_PER_IMG;
  const int sblk = blockIdx.x % BLOCKS_PER_IMG;

  // ---- async global -> LDS staging of this batch's gaussian records ----
  {
    unsigned lbase = (unsigned)(uintptr_t)&sG[0];
    const char* g = (const char*)(ws + (size_t)batch * NGAUSS * STRIDEF);
#pragma unroll
    for (int k = 0; k < LDSBYTES / (256 * 16); ++k) {   /* 9 x b128 per lane */
      unsigned off = (unsigned)((k * 256 + tid) * 16);
      asm volatile("global_load_async_to_lds_b128 %0, %1, off"
                   :: "v"(lbase + off), "v"(g + off) : "memory");
    }
    asm volatile("s_wait_asynccnt 0" ::: "memory");
  }
  __syncthreads();

  const int wave = tid >> 5;
  const int lane = tid & 31;
  const int h = lane >> 4;        /* half: 0 -> tile rows 0-7, 1 -> rows 8-15 */
  const int p = lane & 15;        /* pixel within strip / gaussian for A     */

  const int strip = sblk * WAVES_PER_BLOCK + wave;   /* 0..1599 */
  const int row = strip / (IMG / 16);
  const int colBase = (strip % (IMG / 16)) * 16;

  const float gx = (float)(colBase + p) + 0.5f - 0.5f * (float)IMG;
  const float gy = (float)row + 0.5f - 0.5f * (float)IMG;

  // B operands (pixel features), fixed for the whole gaussian loop.
  // B layout (4x16 f32): VGPR0 = {K-row0 | K-row2}, VGPR1 = {K-row1 | K-row3}.
  v2f B1, B2;
  B1.x = h ? gx * gy : gx * gx;   /* rows 0 / 2 : gx^2, gx*gy */
  B1.y = h ? gx      : gy * gy;   /* rows 1 / 3 : gy^2, gx    */
  B2.x = h ? 0.0f    : gy;        /* rows 4 / 6 : gy,   0     */
  B2.y = h ? 0.0f    : 1.0f;      /* rows 5 / 7 : 1,    0     */

  float T = 1.0f, cR = 0.0f, cG = 0.0f, cB = 0.0f;

  const float* lk = &sG[(size_t)p * STRIDEF + h * 4];  /* A operand source */

  for (int t = 0; t < NGAUSS / 16; ++t) {
    // A operands: gaussian (t*16 + p); low lanes get (K0,K1,K4,K5),
    // high lanes get (K2,K3,0,0) -- one ds_load_b128.
    const float4 kv = *(const float4*)(lk + t * 16 * STRIDEF);
    v2f A1, A2;
    A1.x = kv.x; A1.y = kv.y;
    A2.x = kv.z; A2.y = kv.w;

    v8f acc = {};
    acc = __builtin_amdgcn_wmma_f32_16x16x4_f32(false, A1, false, B1,
                                                (short)0, acc, false, false);
    acc = __builtin_amdgcn_wmma_f32_16x16x4_f32(false, A2, false, B2,
                                                (short)0, acc, false, false);

    // Local ordered compositing over this lane's 8 gaussians (rows h*8..h*8+7)
    float Sr = 0.0f, Sg = 0.0f, Sb = 0.0f, P = 1.0f;
    const int gbase = t * 16 + h * 8;
#pragma unroll
    for (int i = 0; i < 8; ++i) {
      /* alpha = min(op * e^{-q/2}, 0.999), exp output is in [0,1] and
         canonical -> single v_med3_f32 clamp */
      float al = CLAMPF(EXP2F(acc[i]), 0.0f, 0.999f);
      const float4 col = *(const float4*)&sG[(size_t)(gbase + i) * STRIDEF + 8];
      float w = P * al;
      Sr += w * col.x;
      Sg += w * col.y;
      Sb += w * col.z;
      P *= (1.0f - al);
    }

    // Merge the two ordered 8-gaussian segments across the half-wave pair.
    float oSr = __shfl_xor(Sr, 16, 32);
    float oSg = __shfl_xor(Sg, 16, 32);
    float oSb = __shfl_xor(Sb, 16, 32);
    float oP  = __shfl_xor(P, 16, 32);
    float SloR = h ? oSr : Sr, ShiR = h ? Sr : oSr;
    float SloG = h ? oSg : Sg, ShiG = h ? Sg : oSg;
    float SloB = h ? oSb : Sb, ShiB = h ? Sb : oSb;
    float Plo  = h ? oP  : P,  Phi  = h ? P  : oP;

    cR += T * (SloR + Plo * ShiR);
    cG += T * (SloG + Plo * ShiG);
    cB += T * (SloB + Plo * ShiB);
    T *= Plo * Phi;

    if (__all(T < 1e-5f)) break;   /* wave-uniform: remaining contribution ~0 */
  }

  if (h == 0) {
    const int x = colBase + p;
    const size_t base = (size_t)batch * 3 * IMG * IMG + (size_t)row * IMG + x;
    out[base]                 = CLAMPF(cR, 0.0f, 1.0f);
    out[base + IMG * IMG]     = CLAMPF(cG, 0.0f, 1.0f);
    out[base + 2 * IMG * IMG] = CLAMPF(cB, 0.0f, 1.0f);
  }
}

extern "C" void kernel_launch(void* const* d_in, const int* in_sizes, int n_in,
                              void* d_out, int out_size, void* d_ws,
                              size_t ws_size, hipStream_t stream) {
  (void)in_sizes; (void)n_in; (void)out_size; (void)ws_size;
  const float* means  = (const float*)d_in[0];
  const float* scales = (const float*)d_in[1];
  const float* rots   = (const float*)d_in[2];
  const float* colors = (const float*)d_in[3];
  const float* opac   = (const float*)d_in[4];
  float* ws = (float*)d_ws;                 /* 1536 * 12 floats = 73728 B */
  float* out = (float*)d_out;

  gs_prep<<<(NBATCH * NGAUSS + 255) / 256, 256, 0, stream>>>(
      means, scales, rots, colors, opac, ws);
  gs_render<<<NBATCH * BLOCKS_PER_IMG, 256, 0, stream>>>(ws, out);
}